// NMS_77154792505965
// MI455X (gfx1250) — compile-verified
//
#include <hip/hip_runtime.h>
#include <hip/hip_bf16.h>
#include <stdint.h>

typedef float v2f __attribute__((ext_vector_type(2)));
typedef float v8f __attribute__((ext_vector_type(8)));

#define NMS_MAXN 8192

// gfx1250 async global->LDS path, guarded so missing builtins fall back to the
// known-good plain-load staging.
#if __has_builtin(__builtin_amdgcn_global_load_async_to_lds_b128) && \
    __has_builtin(__builtin_amdgcn_s_wait_asynccnt)
#define NMS_ASYNC_LDS 1
// Exact parameter type per hipcc diagnostic: 'vector_size(16) int' pointers in
// addrspace(1) (source) and addrspace(3) (LDS destination).
typedef int nms_v4i __attribute__((vector_size(16)));
typedef __attribute__((address_space(1))) nms_v4i nms_gv4i;
typedef __attribute__((address_space(3))) nms_v4i nms_lv4i;
#else
#define NMS_ASYNC_LDS 0
#endif

// ---------------------------------------------------------------------------
// 1) Sort keys: descending score, ascending index tiebreak (stable argsort).
// ---------------------------------------------------------------------------
__global__ void build_keys_kernel(const float* __restrict__ scores,
                                  unsigned long long* __restrict__ keys, int n) {
    int i = blockIdx.x * blockDim.x + threadIdx.x;
    if (i < n) {
        unsigned u = __float_as_uint(scores[i]);
        // monotone map float -> ascending unsigned
        unsigned s = (u & 0x80000000u) ? ~u : (u | 0x80000000u);
        unsigned hi = ~s; // invert => ascending sort of key gives descending score
        keys[i] = ((unsigned long long)hi << 32) | (unsigned)i;
    }
}

__global__ void bitonic_step_kernel(unsigned long long* __restrict__ keys, int j, int k) {
    int i = blockIdx.x * blockDim.x + threadIdx.x;
    int ixj = i ^ j;
    if (ixj > i) {
        unsigned long long a = keys[i];
        unsigned long long b = keys[ixj];
        bool asc = ((i & k) == 0);
        if ((asc && a > b) || (!asc && a < b)) {
            keys[i] = b;
            keys[ixj] = a;
        }
    }
}

// ---------------------------------------------------------------------------
// 2) Gather boxes into score-descending order.
// ---------------------------------------------------------------------------
__global__ void gather_sorted_kernel(const unsigned long long* __restrict__ keys,
                                     const float4* __restrict__ bboxes,
                                     int* __restrict__ order,
                                     float4* __restrict__ boxes_sorted, int n) {
    int i = blockIdx.x * blockDim.x + threadIdx.x;
    if (i < n) {
        int o = (int)(keys[i] & 0xFFFFFFFFull);
        order[i] = o;
        boxes_sorted[i] = bboxes[o];
    }
}

// ---------------------------------------------------------------------------
// 3) Pairwise IoU suppression bitmask. One wave32 block per 32x32 tile.
//    Column boxes staged in LDS via async global->LDS (ASYNCcnt) when the
//    builtins exist; plain load/store fallback otherwise.
//    mask[r * (n/32) + cb] bit jj set  <=>  IoU(r, cb*32+jj) > thr AND j > r.
// ---------------------------------------------------------------------------
__global__ void iou_mask_kernel(const float4* __restrict__ boxes,
                                const float* __restrict__ thr_p,
                                unsigned* __restrict__ mask, int n) {
    __shared__ float4 colb[32];
    int cb = blockIdx.x;          // column block
    int rb = blockIdx.y;          // row block
    int t  = threadIdx.x;         // 0..31, single wave

#if NMS_ASYNC_LDS
    {
        // addrspace(1) pointer == flat global address; addrspace(3) pointer ==
        // low 32 bits of the flat LDS-aperture address (ISA 10.2 aperture calc).
        nms_gv4i* gsrc = (nms_gv4i*)(uintptr_t)(boxes + (size_t)cb * 32 + t);
        nms_lv4i* ldst = (nms_lv4i*)(unsigned)(uintptr_t)(&colb[t]);
        __builtin_amdgcn_global_load_async_to_lds_b128(gsrc, ldst, 0, 0);
        __builtin_amdgcn_s_wait_asynccnt(0);   // single-wave block: local wait suffices
    }
#else
    colb[t] = boxes[cb * 32 + t];
#endif
    __syncthreads();

    int r = rb * 32 + t;
    __builtin_prefetch(boxes + r, 0, 0);   // global_prefetch_b8
    float4 rB = boxes[r];
    float thr = thr_p[0];
    float rA = (rB.z - rB.x) * (rB.w - rB.y);

    unsigned bits = 0u;
    if (cb * 32 + 31 > r) {        // entire tile strictly below diagonal -> all zero
        #pragma unroll 8
        for (int jj = 0; jj < 32; ++jj) {
            int jglob = cb * 32 + jj;
            float4 cB = colb[jj];
            float iw = fminf(rB.z, cB.z) - fmaxf(rB.x, cB.x);
            float ih = fminf(rB.w, cB.w) - fmaxf(rB.y, cB.y);
            iw = fmaxf(iw, 0.0f);
            ih = fmaxf(ih, 0.0f);
            float inter = iw * ih;
            float cA = (cB.z - cB.x) * (cB.w - cB.y);
            float iou = inter / (rA + cA - inter);
            if ((iou > thr) && (jglob > r)) bits |= (1u << jj);
        }
    }
    mask[(size_t)r * (size_t)(n / 32) + (size_t)cb] = bits;
}

// ---------------------------------------------------------------------------
// 4) Sequential greedy scan (single block, n/32 threads) + stable partition.
//    out_idx / out_valid are aligned with the reference's perm (kept first).
// ---------------------------------------------------------------------------
__global__ void nms_scan_kernel(const unsigned* __restrict__ mask,
                                const int* __restrict__ order,
                                int* __restrict__ out_idx,
                                float* __restrict__ out_valid, int n) {
    __shared__ unsigned remv[NMS_MAXN / 32];
    __shared__ unsigned char keep[NMS_MAXN];
    int t = threadIdx.x;           // 0 .. n/32-1
    int cbk = n / 32;
    remv[t] = 0u;
    __syncthreads();

    for (int i = 0; i < n; ++i) {
        unsigned sup = (remv[i >> 5] >> (i & 31)) & 1u;   // broadcast read
        __syncthreads();                                   // reads before writes
        if (!sup) remv[t] |= mask[(size_t)i * (size_t)cbk + (size_t)t];
        if (t == 0) keep[i] = (unsigned char)(1u - sup);
        __syncthreads();                                   // writes before next read
    }

    if (t == 0) {
        int kc = 0;
        for (int i = 0; i < n; ++i)
            if (keep[i]) { out_idx[kc] = order[i]; out_valid[kc] = 1.0f; ++kc; }
        for (int i = 0; i < n; ++i)
            if (!keep[i]) { out_idx[kc] = order[i]; out_valid[kc] = 0.0f; ++kc; }
    }
}

// ---------------------------------------------------------------------------
// 5) Output: out[p, :] = valid[p] * det[idx[p], :]  ==  diag(valid) x det_g.
//    One wave per 16x16 f32 tile; 4 chained V_WMMA_F32_16X16X4_F32 cover the
//    16x16 diagonal in K=4 slices. Exact in fp32 (0/1 scale, zero-adds).
//    A 16x4 layout (ISA 7.12.2): lanes 0-15 -> M=lane, K={0,1}; lanes 16-31 ->
//    M=lane-16, K={2,3}. D 16x16: VGPR r holds M=r (lanes 0-15) / M=8+r.
// ---------------------------------------------------------------------------
__global__ void scale_gather_wmma_kernel(const float* __restrict__ det,
                                         const int* __restrict__ out_idx,
                                         const float* __restrict__ out_valid,
                                         float* __restrict__ out) {
    int tile = blockIdx.x;          // n/16 tiles of 16 output rows
    int m0   = tile * 16;
    int lane = threadIdx.x;         // 0..31, full wave active (EXEC all 1s)
    int m    = lane & 15;
    int khalf = (lane >= 16) ? 2 : 0;
    int ncol = lane & 15;

    float vm = out_valid[m0 + m];
    v8f acc = {};
    #pragma unroll
    for (int c = 0; c < 4; ++c) {
        int kx = 4 * c + khalf;      // K column held in VGPR0 by this lane
        int ky = kx + 1;             // K column held in VGPR1 by this lane
        v2f A;
        A.x = (m == kx) ? vm : 0.0f; // diag(valid), columns kx/ky
        A.y = (m == ky) ? vm : 0.0f;
        int gx = out_idx[m0 + kx];   // gathered detection rows kx/ky of tile
        int gy = out_idx[m0 + ky];
        v2f B;
        B.x = det[gx * 16 + ncol];
        B.y = det[gy * 16 + ncol];
        acc = __builtin_amdgcn_wmma_f32_16x16x4_f32(
            /*neg_a=*/false, A, /*neg_b=*/false, B,
            /*c_mod=*/(short)0, acc, /*reuse_a=*/false, /*reuse_b=*/false);
    }
    int rowbase = m0 + ((lane >= 16) ? 8 : 0);
    #pragma unroll
    for (int r = 0; r < 8; ++r)
        out[(rowbase + r) * 16 + ncol] = acc[r];
}

// ---------------------------------------------------------------------------
extern "C" void kernel_launch(void* const* d_in, const int* in_sizes, int n_in,
                              void* d_out, int out_size, void* d_ws, size_t ws_size,
                              hipStream_t stream) {
    const float* det    = (const float*)d_in[0];   // (1, N, 16) f32
    const float* bboxes = (const float*)d_in[1];   // (N, 4) f32
    const float* scores = (const float*)d_in[2];   // (N,) f32
    const float* thr    = (const float*)d_in[3];   // scalar f32 (device)

    const int n   = in_sizes[2];   // 8192
    const int cbk = n / 32;        // 256 mask words per row

    // Workspace layout
    char* ws = (char*)d_ws;
    size_t off = 0;
    unsigned long long* keys = (unsigned long long*)(ws + off); off += (size_t)n * 8;
    int*    order        = (int*)(ws + off);                    off += (size_t)n * 4;
    float4* boxes_sorted = (float4*)(ws + off);                 off += (size_t)n * 16;
    unsigned* mask       = (unsigned*)(ws + off);               off += (size_t)n * (size_t)cbk * 4;
    int*    out_idx      = (int*)(ws + off);                    off += (size_t)n * 4;
    float*  out_valid    = (float*)(ws + off);                  off += (size_t)n * 4;

    int blk = 256;
    int grd = (n + blk - 1) / blk;

    // 1) keys + bitonic argsort (descending score, stable)
    build_keys_kernel<<<grd, blk, 0, stream>>>(scores, keys, n);
    for (int k = 2; k <= n; k <<= 1)
        for (int j = k >> 1; j > 0; j >>= 1)
            bitonic_step_kernel<<<grd, blk, 0, stream>>>(keys, j, k);

    // 2) gather boxes into sorted order
    gather_sorted_kernel<<<grd, blk, 0, stream>>>(keys, (const float4*)bboxes,
                                                  order, boxes_sorted, n);

    // 3) pairwise IoU suppression bitmask (one wave32 per 32x32 tile)
    iou_mask_kernel<<<dim3(cbk, n / 32), 32, 0, stream>>>(boxes_sorted, thr, mask, n);

    // 4) sequential greedy scan + stable kept-first partition
    nms_scan_kernel<<<1, cbk, 0, stream>>>(mask, order, out_idx, out_valid, n);

    // 5) WMMA: out = diag(valid) x gathered detections
    scale_gather_wmma_kernel<<<n / 16, 32, 0, stream>>>(det, out_idx, out_valid,
                                                        (float*)d_out);
}